// Shift_act_16484084483761
// MI455X (gfx1250) — compile-verified
//
#include <hip/hip_runtime.h>
#include <hip/hip_bf16.h>
#include <math.h>

#define NUM_CLASSES 1000
#define DIMS        2048
#define NB          4096      // batch (features rows)
#define NR          50000     // reliable bank rows
#define MIN_COUNT   10
#define NPAD        1024      // padded class dim for dist matrix

typedef __attribute__((ext_vector_type(2))) float v2f;
typedef __attribute__((ext_vector_type(4))) float v4f;
typedef __attribute__((ext_vector_type(8))) float v8f;

// ---------------------------------------------------------------------------
// 0. zero the accumulators (must run every call: buffers are not re-poisoned)
// ---------------------------------------------------------------------------
__global__ void zero_ws_kernel(float* sums, float* sumsq, int* counts) {
    int i = blockIdx.x * blockDim.x + threadIdx.x;
    if (i < NUM_CLASSES * DIMS) { sums[i] = 0.0f; sumsq[i] = 0.0f; }
    if (i < NUM_CLASSES) counts[i] = 0;
}

// ---------------------------------------------------------------------------
// 1. per-class counts
// ---------------------------------------------------------------------------
__global__ void count_kernel(const int* labels, int* counts) {
    int i = blockIdx.x * blockDim.x + threadIdx.x;
    if (i < NR) atomicAdd(&counts[labels[i]], 1);
}

// ---------------------------------------------------------------------------
// 2. segment sum / sumsq. Each block owns 4 columns of D, streams all rows,
//    accumulates per-class partials in LDS via ds_add_f32 atomics.
//    Each of the 400MB elements is read exactly once (bandwidth-bound pass).
// ---------------------------------------------------------------------------
__global__ void __launch_bounds__(256) segsum_kernel(const float* __restrict__ rf,
                                                     const int* __restrict__ labels,
                                                     float* __restrict__ sums,
                                                     float* __restrict__ sumsq) {
    __shared__ float ssum[NUM_CLASSES * 4];   // 16 KB
    __shared__ float ssq [NUM_CLASSES * 4];   // 16 KB
    const int tid = threadIdx.x;
    for (int i = tid; i < NUM_CLASSES * 4; i += 256) { ssum[i] = 0.0f; ssq[i] = 0.0f; }
    __syncthreads();

    const int colBase = blockIdx.x * 4;       // gridDim.x == DIMS/4 == 512
    const int cl   = tid & 3;
    const int rOff = tid >> 2;                // 64 rows per iteration
    for (int r0 = 0; r0 < NR; r0 += 64) {
        int r = r0 + rOff;
        if (r < NR) {
            int   lbl = labels[r];
            float v   = rf[(size_t)r * DIMS + colBase + cl];
            atomicAdd(&ssum[lbl * 4 + cl], v);        // ds_add_f32
            atomicAdd(&ssq [lbl * 4 + cl], v * v);    // ds_add_f32
        }
    }
    __syncthreads();
    for (int i = tid; i < NUM_CLASSES * 4; i += 256) {
        int c = i >> 2, cc = i & 3;
        sums [(size_t)c * DIMS + colBase + cc] = ssum[i];
        sumsq[(size_t)c * DIMS + colBase + cc] = ssq[i];
    }
}

// ---------------------------------------------------------------------------
// 3. finalize: mean / unbiased std, fall back to source proto when count<10.
//    Overwrites sums -> mu, sumsq -> std (in place).
// ---------------------------------------------------------------------------
__global__ void finalize_stats_kernel(float* sums, float* sumsq,
                                      const int* __restrict__ counts,
                                      const float* __restrict__ protos) {
    int i = blockIdx.x * blockDim.x + threadIdx.x;
    if (i >= NUM_CLASSES * DIMS) return;
    int   c    = i / DIMS;
    int   cnti = counts[c];
    float cnt  = (float)cnti;
    float mean = sums[i] / fmaxf(cnt, 1.0f);
    float var  = (sumsq[i] - cnt * mean * mean) / fmaxf(cnt - 1.0f, 1.0f);
    float sd   = sqrtf(fmaxf(var, 0.0f));
    if (cnti < MIN_COUNT) { mean = protos[i]; sd = 0.0f; }
    sums[i]  = mean;
    sumsq[i] = sd;
}

// ---------------------------------------------------------------------------
// 4. row sum-of-squares (used for both features and protos)
// ---------------------------------------------------------------------------
__global__ void __launch_bounds__(256) rowsumsq_kernel(const float* __restrict__ src,
                                                       float* __restrict__ out) {
    __shared__ float red[256];
    const int r = blockIdx.x, tid = threadIdx.x;
    float s = 0.0f;
    for (int c = tid; c < DIMS; c += 256) {
        float v = src[(size_t)r * DIMS + c];
        s += v * v;
    }
    red[tid] = s; __syncthreads();
    for (int st = 128; st; st >>= 1) { if (tid < st) red[tid] += red[tid + st]; __syncthreads(); }
    if (tid == 0) out[r] = red[0];
}

// ---------------------------------------------------------------------------
// 5. Gram GEMM G = features @ protos^T via V_WMMA_F32_16X16X4_F32 (full f32),
//    fused with dist = sqrt(max(x2 + y2 - 2G, 0)).
//    Block = 128 threads = 4 waves; each wave computes a 16(M) x 64(N) slab.
//
//    K is processed in chunks of 8 with a legal permutation (same for A & B):
//    lane-half 0 owns physical K {k..k+3}, lane-half 1 owns {k+4..k+7}; the
//    two WMMA steps of a chunk consume (.lo) then (.hi) of ONE b128 load per
//    fragment -> 5 loads feed 8 WMMAs. Register double-buffering (load chunk
//    k while multiplying chunk k-8) keeps loads off the WMMA critical path.
// ---------------------------------------------------------------------------
__global__ void __launch_bounds__(128) gemm_dist_kernel(const float* __restrict__ Fm,
                                                        const float* __restrict__ P,
                                                        const float* __restrict__ x2,
                                                        const float* __restrict__ y2,
                                                        float* __restrict__ dists) {
    const int lane = threadIdx.x & 31;
    const int wave = threadIdx.x >> 5;
    const int M0   = (blockIdx.y * 4 + wave) * 16;   // 256 M-tiles total
    const int C0   = blockIdx.x * 64;                // 16 N-groups (padded to 1024)
    const int lm   = lane & 15;
    const int lh4  = (lane >> 4) << 2;               // 0 or 4: K sub-chunk per lane-half

    v8f acc0 = {}, acc1 = {}, acc2 = {}, acc3 = {};
    const float* arow = Fm + (size_t)(M0 + lm) * DIMS + lh4;

    // clamp out-of-range columns to a valid row; garbage results are never stored
    const int c0r = min(C0 + lm,      NUM_CLASSES - 1);
    const int c1r = min(C0 + lm + 16, NUM_CLASSES - 1);
    const int c2r = min(C0 + lm + 32, NUM_CLASSES - 1);
    const int c3r = min(C0 + lm + 48, NUM_CLASSES - 1);
    const float* b0row = P + (size_t)c0r * DIMS + lh4;
    const float* b1row = P + (size_t)c1r * DIMS + lh4;
    const float* b2row = P + (size_t)c2r * DIMS + lh4;
    const float* b3row = P + (size_t)c3r * DIMS + lh4;

    // prologue: chunk 0
    v4f a  = *(const v4f*)(arow);
    v4f b0 = *(const v4f*)(b0row);
    v4f b1 = *(const v4f*)(b1row);
    v4f b2 = *(const v4f*)(b2row);
    v4f b3 = *(const v4f*)(b3row);

    #pragma unroll 2
    for (int k = 8; k < DIMS; k += 8) {
        // issue next chunk's loads first (double buffer)
        __builtin_prefetch(arow + k + 64, 0, 3);     // global_prefetch_b8, lines ahead
        v4f an  = *(const v4f*)(arow  + k);
        v4f b0n = *(const v4f*)(b0row + k);
        v4f b1n = *(const v4f*)(b1row + k);
        v4f b2n = *(const v4f*)(b2row + k);
        v4f b3n = *(const v4f*)(b3row + k);

        v2f alo = __builtin_shufflevector(a, a, 0, 1);
        v2f ahi = __builtin_shufflevector(a, a, 2, 3);
        acc0 = __builtin_amdgcn_wmma_f32_16x16x4_f32(false, alo, false, __builtin_shufflevector(b0, b0, 0, 1), (short)0, acc0, false, false);
        acc1 = __builtin_amdgcn_wmma_f32_16x16x4_f32(false, alo, false, __builtin_shufflevector(b1, b1, 0, 1), (short)0, acc1, false, false);
        acc2 = __builtin_amdgcn_wmma_f32_16x16x4_f32(false, alo, false, __builtin_shufflevector(b2, b2, 0, 1), (short)0, acc2, false, false);
        acc3 = __builtin_amdgcn_wmma_f32_16x16x4_f32(false, alo, false, __builtin_shufflevector(b3, b3, 0, 1), (short)0, acc3, false, false);
        acc0 = __builtin_amdgcn_wmma_f32_16x16x4_f32(false, ahi, false, __builtin_shufflevector(b0, b0, 2, 3), (short)0, acc0, false, false);
        acc1 = __builtin_amdgcn_wmma_f32_16x16x4_f32(false, ahi, false, __builtin_shufflevector(b1, b1, 2, 3), (short)0, acc1, false, false);
        acc2 = __builtin_amdgcn_wmma_f32_16x16x4_f32(false, ahi, false, __builtin_shufflevector(b2, b2, 2, 3), (short)0, acc2, false, false);
        acc3 = __builtin_amdgcn_wmma_f32_16x16x4_f32(false, ahi, false, __builtin_shufflevector(b3, b3, 2, 3), (short)0, acc3, false, false);

        a = an; b0 = b0n; b1 = b1n; b2 = b2n; b3 = b3n;
    }

    // epilogue: last chunk
    {
        v2f alo = __builtin_shufflevector(a, a, 0, 1);
        v2f ahi = __builtin_shufflevector(a, a, 2, 3);
        acc0 = __builtin_amdgcn_wmma_f32_16x16x4_f32(false, alo, false, __builtin_shufflevector(b0, b0, 0, 1), (short)0, acc0, false, false);
        acc1 = __builtin_amdgcn_wmma_f32_16x16x4_f32(false, alo, false, __builtin_shufflevector(b1, b1, 0, 1), (short)0, acc1, false, false);
        acc2 = __builtin_amdgcn_wmma_f32_16x16x4_f32(false, alo, false, __builtin_shufflevector(b2, b2, 0, 1), (short)0, acc2, false, false);
        acc3 = __builtin_amdgcn_wmma_f32_16x16x4_f32(false, alo, false, __builtin_shufflevector(b3, b3, 0, 1), (short)0, acc3, false, false);
        acc0 = __builtin_amdgcn_wmma_f32_16x16x4_f32(false, ahi, false, __builtin_shufflevector(b0, b0, 2, 3), (short)0, acc0, false, false);
        acc1 = __builtin_amdgcn_wmma_f32_16x16x4_f32(false, ahi, false, __builtin_shufflevector(b1, b1, 2, 3), (short)0, acc1, false, false);
        acc2 = __builtin_amdgcn_wmma_f32_16x16x4_f32(false, ahi, false, __builtin_shufflevector(b2, b2, 2, 3), (short)0, acc2, false, false);
        acc3 = __builtin_amdgcn_wmma_f32_16x16x4_f32(false, ahi, false, __builtin_shufflevector(b3, b3, 2, 3), (short)0, acc3, false, false);
    }

    // C/D layout: VGPR v holds M = v (lanes 0-15) or M = v+8 (lanes 16-31), N = lane&15
    const int mhi = (lane < 16) ? 0 : 8;
    for (int v = 0; v < 8; ++v) {
        int   row = M0 + v + mhi;
        float xr  = x2[row];
        float g[4] = { acc0[v], acc1[v], acc2[v], acc3[v] };
        #pragma unroll
        for (int nt = 0; nt < 4; ++nt) {
            int col = C0 + nt * 16 + lm;
            if (col < NUM_CLASSES) {
                float d2 = xr + y2[col] - 2.0f * g[nt];
                dists[(size_t)row * NPAD + col] = sqrtf(fmaxf(d2, 0.0f));
            }
        }
    }
}

// ---------------------------------------------------------------------------
// 6. top-3 smallest per row: one wave32 per row, per-lane top3 + LDS merge
// ---------------------------------------------------------------------------
__global__ void __launch_bounds__(32) top3_kernel(const float* __restrict__ dists,
                                                  int* __restrict__ cand) {
    __shared__ float sv[96];
    __shared__ int   si[96];
    const int r = blockIdx.x, lane = threadIdx.x;
    float v0 = 3.4e38f, v1 = 3.4e38f, v2 = 3.4e38f;
    int   i0 = 0, i1 = 0, i2 = 0;
    for (int c = lane; c < NUM_CLASSES; c += 32) {
        float d = dists[(size_t)r * NPAD + c];
        if (d < v0)      { v2 = v1; i2 = i1; v1 = v0; i1 = i0; v0 = d; i0 = c; }
        else if (d < v1) { v2 = v1; i2 = i1; v1 = d;  i1 = c; }
        else if (d < v2) { v2 = d;  i2 = c; }
    }
    sv[lane * 3 + 0] = v0; si[lane * 3 + 0] = i0;
    sv[lane * 3 + 1] = v1; si[lane * 3 + 1] = i1;
    sv[lane * 3 + 2] = v2; si[lane * 3 + 2] = i2;
    __syncthreads();
    if (lane == 0) {
        float b0 = 3.4e38f, b1 = 3.4e38f, b2 = 3.4e38f;
        int   j0 = 0, j1 = 0, j2 = 0;
        for (int t = 0; t < 96; ++t) {
            float d = sv[t]; int ix = si[t];
            if (d < b0)      { b2 = b1; j2 = j1; b1 = b0; j1 = j0; b0 = d; j0 = ix; }
            else if (d < b1) { b2 = b1; j2 = j1; b1 = d;  j1 = ix; }
            else if (d < b2) { b2 = d;  j2 = ix; }
        }
        cand[r * 3 + 0] = j0; cand[r * 3 + 1] = j1; cand[r * 3 + 2] = j2;
    }
}

// ---------------------------------------------------------------------------
// 7. per-row PCL loss: Sum diff^2 per candidate -> maha = nrm/max(nrm,eps),
//    then -log(exp(-min)/sum exp(-maha))
// ---------------------------------------------------------------------------
__global__ void __launch_bounds__(256) rowloss_kernel(const float* __restrict__ Fm,
                                                      const float* __restrict__ mu,
                                                      const float* __restrict__ sd,
                                                      const int* __restrict__ cand,
                                                      float* __restrict__ rloss) {
    __shared__ float red[3 * 256];
    const int r = blockIdx.x, tid = threadIdx.x;
    const int c0 = cand[r * 3 + 0], c1 = cand[r * 3 + 1], c2 = cand[r * 3 + 2];
    float p0 = 0.0f, p1 = 0.0f, p2 = 0.0f;
    for (int d = tid; d < DIMS; d += 256) {
        float f  = Fm[(size_t)r * DIMS + d];
        float d0 = (f - mu[(size_t)c0 * DIMS + d]) / (sd[(size_t)c0 * DIMS + d] + 1e-3f);
        float d1 = (f - mu[(size_t)c1 * DIMS + d]) / (sd[(size_t)c1 * DIMS + d] + 1e-3f);
        float d2 = (f - mu[(size_t)c2 * DIMS + d]) / (sd[(size_t)c2 * DIMS + d] + 1e-3f);
        p0 += d0 * d0; p1 += d1 * d1; p2 += d2 * d2;
    }
    red[tid] = p0; red[256 + tid] = p1; red[512 + tid] = p2;
    __syncthreads();
    for (int s = 128; s; s >>= 1) {
        if (tid < s) {
            red[tid]       += red[tid + s];
            red[256 + tid] += red[256 + tid + s];
            red[512 + tid] += red[512 + tid + s];
        }
        __syncthreads();
    }
    if (tid == 0) {
        float n0 = sqrtf(red[0]), n1 = sqrtf(red[256]), n2 = sqrtf(red[512]);
        float h0 = n0 / fmaxf(n0, 1e-12f);
        float h1 = n1 / fmaxf(n1, 1e-12f);
        float h2 = n2 / fmaxf(n2, 1e-12f);
        float mn   = fminf(h0, fminf(h1, h2));
        float ssum = expf(-h0) + expf(-h1) + expf(-h2);
        rloss[r] = -logf(expf(-mn) / ssum);
    }
}

// ---------------------------------------------------------------------------
// 8. final mean over 4096 rows
// ---------------------------------------------------------------------------
__global__ void __launch_bounds__(256) reduce_loss_kernel(const float* __restrict__ rloss,
                                                          float* __restrict__ out) {
    __shared__ float red[256];
    const int tid = threadIdx.x;
    float s = 0.0f;
    for (int i = tid; i < NB; i += 256) s += rloss[i];
    red[tid] = s; __syncthreads();
    for (int st = 128; st; st >>= 1) { if (tid < st) red[tid] += red[tid + st]; __syncthreads(); }
    if (tid == 0) out[0] = red[0] / (float)NB;
}

// ---------------------------------------------------------------------------
extern "C" void kernel_launch(void* const* d_in, const int* in_sizes, int n_in,
                              void* d_out, int out_size, void* d_ws, size_t ws_size,
                              hipStream_t stream) {
    const float* features = (const float*)d_in[0];  // [4096, 2048]
    const float* rf       = (const float*)d_in[1];  // [50000, 2048]
    const float* protos   = (const float*)d_in[2];  // [1000, 2048]
    const int*   labels   = (const int*)d_in[3];    // [50000]

    // workspace layout (floats unless noted), ~33 MB total
    float* sums   = (float*)d_ws;                              // [1000*2048] -> mu (in place)
    float* sumsq  = sums + (size_t)NUM_CLASSES * DIMS;         // [1000*2048] -> std (in place)
    int*   counts = (int*)(sumsq + (size_t)NUM_CLASSES * DIMS);// [1000] (padded 1024)
    float* x2     = (float*)(counts + 1024);                   // [4096]
    float* y2     = x2 + NB;                                   // [1024 pad]
    float* dists  = y2 + NPAD;                                 // [4096*1024]
    int*   cand   = (int*)(dists + (size_t)NB * NPAD);         // [4096*3] (padded 4)
    float* rloss  = (float*)(cand + NB * 4);                   // [4096]

    zero_ws_kernel<<<(NUM_CLASSES * DIMS + 255) / 256, 256, 0, stream>>>(sums, sumsq, counts);
    count_kernel<<<(NR + 255) / 256, 256, 0, stream>>>(labels, counts);
    segsum_kernel<<<DIMS / 4, 256, 0, stream>>>(rf, labels, sums, sumsq);
    finalize_stats_kernel<<<(NUM_CLASSES * DIMS + 255) / 256, 256, 0, stream>>>(sums, sumsq, counts, protos);
    rowsumsq_kernel<<<NB, 256, 0, stream>>>(features, x2);
    rowsumsq_kernel<<<NUM_CLASSES, 256, 0, stream>>>(protos, y2);
    dim3 gg(NPAD / 64, NB / 64);   // 16 x 64 blocks, 4 waves each
    gemm_dist_kernel<<<gg, 128, 0, stream>>>(features, protos, x2, y2, dists);
    top3_kernel<<<NB, 32, 0, stream>>>(dists, cand);
    rowloss_kernel<<<NB, 256, 0, stream>>>(features, sums, sumsq, cand, rloss);
    reduce_loss_kernel<<<1, 256, 0, stream>>>(rloss, (float*)d_out);
}